// MultiHeadAttention_4544075399604
// MI455X (gfx1250) — compile-verified
//
#include <hip/hip_runtime.h>
#include <hip/hip_bf16.h>

// ---------------------------------------------------------------------------
// Multi-head causal attention, fp32 end-to-end on the CDNA5 WMMA fp32 pipe
// (v_wmma_f32_16x16x4_f32), async global->LDS staging (ASYNCcnt), double-
// buffered LDS tiles, and pair-interleaved B-matrix layouts so every WMMA
// B fragment is a single ds_load_b64. b=2, s=2048, d=1024, h=16, hd=64.
// ---------------------------------------------------------------------------

typedef __attribute__((ext_vector_type(2))) float v2f;
typedef __attribute__((ext_vector_type(8))) float v8f;

#define WMMA_F32(a, b, c) \
    __builtin_amdgcn_wmma_f32_16x16x4_f32(false, (a), false, (b), (short)0, (c), false, false)

#define B_   2
#define S_   2048
#define D_   1024
#define H_   16
#define HD_  64
#define BS_  (B_ * S_)   // 4096

// LDS row strides (floats): byte stride multiple of 16 (async b128 dests)
// and bank-conflict-free fragment reads.
#define LDA  36   // 32-wide A tile           (144 B rows)
#define LDB2 264  // 256-wide packed B tile   (1056 B rows, 16 k-pair rows)
#define LDKV 68   // 64-wide K tile           (272 B rows)
#define LDV2 136  // 128-wide packed V tile   (544 B rows, 8 key-pair rows)
#define LDP  18   // 16-wide P staging tile

// Async global->LDS 16-byte copy (GLOBAL_LOAD_ASYNC_TO_LDS_B128, ASYNCcnt).
__device__ __forceinline__ void async_ld16(const void* g, const void* lds) {
    unsigned loff = (unsigned)(unsigned long long)lds;  // low 32 bits of the pointer give the LDS address
    unsigned long long ga = (unsigned long long)g;
    asm volatile("global_load_async_to_lds_b128 %0, %1, off"
                 :: "v"(loff), "v"(ga) : "memory");
}
#define WAIT_ASYNC(imm) asm volatile("s_wait_asynccnt " #imm ::: "memory")

// ---------------------------------------------------------------------------
// Kernel 0: pack the four weight matrices into pair-interleaved layout:
// P[(k>>1)*D + n)*2 + (k&1)] = W[k*D + n]. grid = (2048, 4), block = 256.
// ---------------------------------------------------------------------------
__global__ __launch_bounds__(256) void mha_pack_w(
    const float* __restrict__ Wq, const float* __restrict__ Wk,
    const float* __restrict__ Wv, const float* __restrict__ Wo,
    float* __restrict__ Pq, float* __restrict__ Pk,
    float* __restrict__ Pv, float* __restrict__ Po)
{
    const float* src = (blockIdx.y == 0) ? Wq : (blockIdx.y == 1) ? Wk
                     : (blockIdx.y == 2) ? Wv : Wo;
    float*       dst = (blockIdx.y == 0) ? Pq : (blockIdx.y == 1) ? Pk
                     : (blockIdx.y == 2) ? Pv : Po;
    const int idx = blockIdx.x * 256 + threadIdx.x;  // 0 .. 512*1024-1
    const int kp = idx >> 10, n = idx & (D_ - 1);
    v2f w;
    w.x = src[(size_t)(2 * kp + 0) * D_ + n];
    w.y = src[(size_t)(2 * kp + 1) * D_ + n];
    *(v2f*)(&dst[((size_t)kp * D_ + n) * 2]) = w;
}

// ---------------------------------------------------------------------------
// Kernel 1: QKV projection from packed weights. grid = (64, 8, 3),
// block = 256 (8 waves as 2x4; each wave computes a 32x32 = 2x2 WMMA block).
// Q,K written [b,h,s,hd]; V written pair-interleaved over keys.
// ---------------------------------------------------------------------------
__global__ __launch_bounds__(256) void mha_qkv_gemm(
    const float* __restrict__ x,
    const float* __restrict__ Pq, const float* __restrict__ Pk,
    const float* __restrict__ Pv,
    float* __restrict__ Qo, float* __restrict__ Ko, float* __restrict__ Vo)
{
    __shared__ float Asm[2][64 * LDA];
    __shared__ float Bsm[2][16 * LDB2];

    const int tid  = threadIdx.x;
    const int wave = tid >> 5;
    const int lane = tid & 31;
    const int lg   = lane >> 4;
    const int lr   = lane & 15;
    const int wm   = wave >> 2;    // 0..1 : 32-row sub-tile
    const int wn   = wave & 3;     // 0..3 : 32-col sub-tile

    const int row0 = blockIdx.x * 64;
    const int col0 = blockIdx.y * 128;

    const float* W    = (blockIdx.z == 0) ? Pq : (blockIdx.z == 1 ? Pk : Pv);
    float*       Outp = (blockIdx.z == 0) ? Qo : (blockIdx.z == 1 ? Ko : Vo);
    const bool   isV  = (blockIdx.z == 2);

    // A: 64x32 (2 b128/thread). B packed: 16 pair-rows x 256 (4 b128/thread).
    auto stage = [&](int k0, int buf) {
#pragma unroll
        for (int j = 0; j < 2; ++j) {
            const int idx = tid + j * 256;
            const int r = idx >> 3, c = (idx & 7) * 4;
            async_ld16(x + (size_t)(row0 + r) * D_ + k0 + c, &Asm[buf][r * LDA + c]);
        }
#pragma unroll
        for (int j = 0; j < 4; ++j) {
            const int idx = tid + j * 256;
            const int r = idx >> 6, c = (idx & 63) * 4;   // 16 rows x 64 float4
            async_ld16(W + ((size_t)(k0 / 2) + r) * (2 * D_) + col0 * 2 + c,
                       &Bsm[buf][r * LDB2 + c]);
        }
    };

    v8f acc00 = {}, acc01 = {}, acc10 = {}, acc11 = {};

    stage(0, 0);
    for (int kc = 0; kc < D_ / 32; ++kc) {
        const int buf = kc & 1;
        if (kc + 1 < D_ / 32) {
            stage((kc + 1) * 32, 1 - buf);  // prefetch next chunk
            WAIT_ASYNC(0x6);
        } else {
            WAIT_ASYNC(0x0);
        }
        __syncthreads();

#pragma unroll
        for (int ks = 0; ks < 32; ks += 4) {
            const int kr = ks + 2 * lg;
            v2f a0 = *(const v2f*)(&Asm[buf][(wm * 32 + lr) * LDA + kr]);
            v2f a1 = *(const v2f*)(&Asm[buf][(wm * 32 + 16 + lr) * LDA + kr]);
            const int bi = ((ks >> 1) + lg) * LDB2 + (wn * 32 + lr) * 2;
            v2f b0 = *(const v2f*)(&Bsm[buf][bi]);       // (kr,n),(kr+1,n)
            v2f b1 = *(const v2f*)(&Bsm[buf][bi + 32]);  // n + 16
            acc00 = WMMA_F32(a0, b0, acc00);
            acc01 = WMMA_F32(a0, b1, acc01);
            acc10 = WMMA_F32(a1, b0, acc10);
            acc11 = WMMA_F32(a1, b1, acc11);
        }
        __syncthreads();
    }

    // Store C (C-layout: reg r -> row r + 8*lg, col lr).
#pragma unroll
    for (int mt = 0; mt < 2; ++mt) {
#pragma unroll
        for (int nt = 0; nt < 2; ++nt) {
            const v8f& a = mt ? (nt ? acc11 : acc10) : (nt ? acc01 : acc00);
#pragma unroll
            for (int r = 0; r < 8; ++r) {
                const int m    = row0 + wm * 32 + mt * 16 + r + 8 * lg;
                const int n    = col0 + wn * 32 + nt * 16 + lr;
                const int bb   = m >> 11, srow = m & (S_ - 1);
                const int head = n >> 6,  hc   = n & (HD_ - 1);
                size_t o;
                if (isV)  // pair-interleaved over keys for WMMA-B consumption
                    o = ((size_t)(bb * H_ + head) * (S_ / 2) + (srow >> 1)) * (2 * HD_)
                        + hc * 2 + (srow & 1);
                else
                    o = ((size_t)(bb * H_ + head) * S_ + srow) * HD_ + hc;
                Outp[o] = a[r];
            }
        }
    }
}

// ---------------------------------------------------------------------------
// Kernel 2: causal flash attention. grid = (32, 32), block = 128 (4 waves,
// each owns a 16-query tile; double-buffered async K/V LDS tiles).
// ---------------------------------------------------------------------------
__global__ __launch_bounds__(128) void mha_attn_fwd(
    const float* __restrict__ Qw, const float* __restrict__ Kw,
    const float* __restrict__ Vw, float* __restrict__ ctx)
{
    __shared__ float Ksm[2][16 * LDKV];
    __shared__ float Vsm[2][8 * LDV2];
    __shared__ float Psm[4][16 * LDP];

    const int tid  = threadIdx.x;
    const int wave = tid >> 5;
    const int lane = tid & 31;
    const int lg   = lane >> 4;
    const int lr   = lane & 15;

    const int bh = blockIdx.y;                   // b*16 + h
    const int q0 = blockIdx.x * 64 + wave * 16;  // this wave's query base

    const float* Qb = Qw + (size_t)bh * S_ * HD_;
    const float* Kb = Kw + (size_t)bh * S_ * HD_;
    const float* Vb = Vw + (size_t)bh * S_ * HD_;  // packed: [kpair][dim*2+par]

    // Async-stage one 16x64 K tile + packed V tile: 2+2 b128 per thread.
    auto stage = [&](int kt, int buf) {
#pragma unroll
        for (int j = 0; j < 2; ++j) {
            const int idx = tid + j * 128;                 // 0..255
            const int kr = idx >> 4, kc = (idx & 15) * 4;  // 16 rows x 16 f4
            async_ld16(Kb + (size_t)(kt * 16 + kr) * HD_ + kc,
                       &Ksm[buf][kr * LDKV + kc]);
            const int vr = idx >> 5, vc = (idx & 31) * 4;  // 8 rows x 32 f4
            async_ld16(Vb + (size_t)(kt * 8 + vr) * (2 * HD_) + vc,
                       &Vsm[buf][vr * LDV2 + vc]);
        }
    };

    // Q fragments for all 16 K-steps, pre-scaled by 1/sqrt(hd).
    const float qscale = 0.125f;
    v2f qf[16];
#pragma unroll
    for (int i = 0; i < 16; ++i) {
        const float* p = Qb + (size_t)(q0 + lr) * HD_ + i * 4 + 2 * lg;
        qf[i].x = p[0] * qscale;
        qf[i].y = p[1] * qscale;
    }

    v8f acc0 = {}, acc1 = {}, acc2 = {}, acc3 = {};
    float mrow[8], lsum[8];
#pragma unroll
    for (int r = 0; r < 8; ++r) { mrow[r] = -1e30f; lsum[r] = 0.0f; }

    const int ktiles = blockIdx.x * 4 + 4;  // causal extent of whole block
    stage(0, 0);
    for (int kt = 0; kt < ktiles; ++kt) {
        const int buf = kt & 1;
        if (kt + 1 < ktiles) {
            stage(kt + 1, 1 - buf);  // prefetch next K/V tile
            WAIT_ASYNC(0x4);
        } else {
            WAIT_ASYNC(0x0);
        }
        __syncthreads();

        // Scores S = (Q/sqrt(hd)) @ K^T : 16 WMMA K-steps over hd=64.
        v8f s = {};
#pragma unroll
        for (int i = 0; i < 16; ++i) {
            v2f b = *(const v2f*)(&Ksm[buf][lr * LDKV + i * 4 + 2 * lg]);
            s = WMMA_F32(qf[i], b, s);
        }

        // Causal mask (C layout: reg r -> query q0+r+8*lg, key kt*16+lr).
        const int kg = kt * 16 + lr;
        float mnew[8], alpha[8];
#pragma unroll
        for (int r = 0; r < 8; ++r) {
            const int qg = q0 + r + 8 * lg;
            if (kg > qg) s[r] = -1e30f;
            float v = s[r];  // row max over the 16 lanes of this half-wave
            v = fmaxf(v, __shfl_xor(v, 1, 32));
            v = fmaxf(v, __shfl_xor(v, 2, 32));
            v = fmaxf(v, __shfl_xor(v, 4, 32));
            v = fmaxf(v, __shfl_xor(v, 8, 32));
            mnew[r]  = fmaxf(mrow[r], v);
            alpha[r] = __expf(mrow[r] - mnew[r]);
            mrow[r]  = mnew[r];
        }
#pragma unroll
        for (int r = 0; r < 8; ++r) {
            const float p = __expf(s[r] - mnew[r]);
            s[r] = p;
            float t = p;
            t += __shfl_xor(t, 1, 32);
            t += __shfl_xor(t, 2, 32);
            t += __shfl_xor(t, 4, 32);
            t += __shfl_xor(t, 8, 32);
            lsum[r] = lsum[r] * alpha[r] + t;
            acc0[r] *= alpha[r];
            acc1[r] *= alpha[r];
            acc2[r] *= alpha[r];
            acc3[r] *= alpha[r];
        }

        // Transpose P (C layout) -> A-fragment layout via per-wave LDS tile.
        float* Pst = &Psm[wave][0];
#pragma unroll
        for (int r = 0; r < 8; ++r)
            Pst[(r + 8 * lg) * LDP + lr] = s[r];
        __syncthreads();

        // ctx += P @ V : 4 N-tiles of 16, 4 K-steps each; packed V -> b64 frags.
#pragma unroll
        for (int ks = 0; ks < 16; ks += 4) {
            v2f a = *(const v2f*)(&Pst[lr * LDP + ks + 2 * lg]);
            const int vr = ((ks >> 1) + lg) * LDV2;
            v2f b0 = *(const v2f*)(&Vsm[buf][vr + ( 0 + lr) * 2]);
            v2f b1 = *(const v2f*)(&Vsm[buf][vr + (16 + lr) * 2]);
            v2f b2 = *(const v2f*)(&Vsm[buf][vr + (32 + lr) * 2]);
            v2f b3 = *(const v2f*)(&Vsm[buf][vr + (48 + lr) * 2]);
            acc0 = WMMA_F32(a, b0, acc0);
            acc1 = WMMA_F32(a, b1, acc1);
            acc2 = WMMA_F32(a, b2, acc2);
            acc3 = WMMA_F32(a, b3, acc3);
        }
        __syncthreads();  // all reads of buf done before it is re-staged
    }

    // Normalize and store ctx in [b, s, d] layout for the output projection.
    const int bb = bh >> 4, head = bh & 15;
#pragma unroll
    for (int r = 0; r < 8; ++r) {
        const float inv = 1.0f / lsum[r];
        const int m = q0 + r + 8 * lg;
        float* dst = ctx + (size_t)(bb * S_ + m) * D_ + head * HD_;
        dst[ 0 + lr] = acc0[r] * inv;
        dst[16 + lr] = acc1[r] * inv;
        dst[32 + lr] = acc2[r] * inv;
        dst[48 + lr] = acc3[r] * inv;
    }
}

// ---------------------------------------------------------------------------
// Kernel 3: output projection from packed Wo. out = ctx @ Wo + bo.
// grid = (64, 8), block = 256, 2x2 WMMA register blocking per wave.
// ---------------------------------------------------------------------------
__global__ __launch_bounds__(256) void mha_oproj_gemm(
    const float* __restrict__ ctx, const float* __restrict__ Po,
    const float* __restrict__ bo, float* __restrict__ out)
{
    __shared__ float Asm[2][64 * LDA];
    __shared__ float Bsm[2][16 * LDB2];

    const int tid  = threadIdx.x;
    const int wave = tid >> 5;
    const int lane = tid & 31;
    const int lg   = lane >> 4;
    const int lr   = lane & 15;
    const int wm   = wave >> 2;
    const int wn   = wave & 3;

    const int row0 = blockIdx.x * 64;
    const int col0 = blockIdx.y * 128;

    auto stage = [&](int k0, int buf) {
#pragma unroll
        for (int j = 0; j < 2; ++j) {
            const int idx = tid + j * 256;
            const int r = idx >> 3, c = (idx & 7) * 4;
            async_ld16(ctx + (size_t)(row0 + r) * D_ + k0 + c, &Asm[buf][r * LDA + c]);
        }
#pragma unroll
        for (int j = 0; j < 4; ++j) {
            const int idx = tid + j * 256;
            const int r = idx >> 6, c = (idx & 63) * 4;
            async_ld16(Po + ((size_t)(k0 / 2) + r) * (2 * D_) + col0 * 2 + c,
                       &Bsm[buf][r * LDB2 + c]);
        }
    };

    v8f acc00 = {}, acc01 = {}, acc10 = {}, acc11 = {};

    stage(0, 0);
    for (int kc = 0; kc < D_ / 32; ++kc) {
        const int buf = kc & 1;
        if (kc + 1 < D_ / 32) {
            stage((kc + 1) * 32, 1 - buf);
            WAIT_ASYNC(0x6);
        } else {
            WAIT_ASYNC(0x0);
        }
        __syncthreads();

#pragma unroll
        for (int ks = 0; ks < 32; ks += 4) {
            const int kr = ks + 2 * lg;
            v2f a0 = *(const v2f*)(&Asm[buf][(wm * 32 + lr) * LDA + kr]);
            v2f a1 = *(const v2f*)(&Asm[buf][(wm * 32 + 16 + lr) * LDA + kr]);
            const int bi = ((ks >> 1) + lg) * LDB2 + (wn * 32 + lr) * 2;
            v2f b0 = *(const v2f*)(&Bsm[buf][bi]);
            v2f b1 = *(const v2f*)(&Bsm[buf][bi + 32]);
            acc00 = WMMA_F32(a0, b0, acc00);
            acc01 = WMMA_F32(a0, b1, acc01);
            acc10 = WMMA_F32(a1, b0, acc10);
            acc11 = WMMA_F32(a1, b1, acc11);
        }
        __syncthreads();
    }

#pragma unroll
    for (int mt = 0; mt < 2; ++mt) {
#pragma unroll
        for (int nt = 0; nt < 2; ++nt) {
            const v8f& a = mt ? (nt ? acc11 : acc10) : (nt ? acc01 : acc00);
            const int n  = col0 + wn * 32 + nt * 16 + lr;
            const float bias = bo[n];
#pragma unroll
            for (int r = 0; r < 8; ++r) {
                const int m = row0 + wm * 32 + mt * 16 + r + 8 * lg;
                out[(size_t)m * D_ + n] = a[r] + bias;
            }
        }
    }
}

// ---------------------------------------------------------------------------
extern "C" void kernel_launch(void* const* d_in, const int* in_sizes, int n_in,
                              void* d_out, int out_size, void* d_ws, size_t ws_size,
                              hipStream_t stream)
{
    const float* x  = (const float*)d_in[0];
    const float* Wq = (const float*)d_in[1];
    const float* Wk = (const float*)d_in[2];
    const float* Wv = (const float*)d_in[3];
    const float* Wo = (const float*)d_in[4];
    const float* bo = (const float*)d_in[5];
    float* out = (float*)d_out;

    // Workspace: Q,K [b,h,s,hd], V [b,h,s/2,hd,2] + ctx [b,s,d] (16 MB each) + 4 packed
    // weight matrices (4 MB each) = 80 MB total.
    const size_t elems = (size_t)BS_ * D_;  // 4 Mi floats
    const size_t welems = (size_t)D_ * D_;  // 1 Mi floats
    float* Qw = (float*)d_ws;
    float* Kw = Qw + elems;
    float* Vw = Kw + elems;
    float* Cw = Vw + elems;
    float* Pq = Cw + elems;
    float* Pk = Pq + welems;
    float* Pv = Pk + welems;
    float* Po = Pv + welems;

    dim3 gPack((D_ / 2) * D_ / 256, 4);
    mha_pack_w<<<gPack, 256, 0, stream>>>(Wq, Wk, Wv, Wo, Pq, Pk, Pv, Po);

    dim3 gQKV(BS_ / 64, D_ / 128, 3);
    mha_qkv_gemm<<<gQKV, 256, 0, stream>>>(x, Pq, Pk, Pv, Qw, Kw, Vw);

    dim3 gAtt(S_ / 64, B_ * H_);
    mha_attn_fwd<<<gAtt, 128, 0, stream>>>(Qw, Kw, Vw, Cw);

    dim3 gO(BS_ / 64, D_ / 128);
    mha_oproj_gemm<<<gO, 256, 0, stream>>>(Cw, Po, bo, out);
}